// CorrelationCycleLoss_14001593385464
// MI455X (gfx1250) — compile-verified
//
#include <hip/hip_runtime.h>

typedef __attribute__((ext_vector_type(2))) float v2f;
typedef __attribute__((ext_vector_type(8))) float v8f;

#define NB 4
#define NN 4096
#define ND 128
#define ROWS_PER_BLOCK 64
#define JT 32
#define YSTRIDE 132               // 128 + 4 floats pad -> LDS bank spread for ds_load_b64
#define CLIP_HI_BITS 0x42C80000u  // bits of 100.0f (P is clipped to [0,100])

// ---------------- row squared-norms: out[r] = sum_d p[r][d]^2 ----------------
__global__ void row_norms_kernel(const float* __restrict__ p, float* __restrict__ out,
                                 int rows, int d) {
    int r = blockIdx.x * blockDim.x + threadIdx.x;
    if (r >= rows) return;
    const float4* pr = (const float4*)(p + (size_t)r * d);
    float s = 0.0f;
    #pragma unroll 8
    for (int q = 0; q < d / 4; ++q) {
        float4 v = pr[q];
        s += v.x * v.x + v.y * v.y + v.z * v.z + v.w * v.w;
    }
    out[r] = s;
}

// ---------------- init global colmin to bits(100.0f) ----------------
__global__ void init_colmin_kernel(unsigned* __restrict__ colmin, int n) {
    int i = blockIdx.x * blockDim.x + threadIdx.x;
    if (i < n) colmin[i] = CLIP_HI_BITS;
}

// Issue this thread's share (8x) of async B128 copies of y[j0..j0+31][0..127]
// into the given LDS buffer. Tracked by ASYNCcnt; completes in issue order.
__device__ __forceinline__ void stage_tile_async(float* sbuf, const float* __restrict__ yb,
                                                 int j0, int tid) {
    #pragma unroll
    for (int u = 0; u < (JT * ND / 4) / 128; ++u) {      // 8 iterations
        int idx = tid + u * 128;
        int row = idx >> 5;                               // ND/4 = 32 float4 per row
        int q   = idx & 31;
        unsigned laddr = (unsigned)(uintptr_t)(sbuf + (row * YSTRIDE + q * 4));
        const float* g = yb + (size_t)(j0 + row) * ND + q * 4;
        asm volatile("global_load_async_to_lds_b128 %0, %1, off"
                     :: "v"(laddr), "v"(g) : "memory");
    }
}

// ---------------- fused chamfer GEMM + clip + min ----------------
// zz[b,i,j] = x_i . y_j via V_WMMA_F32_16X16X4_F32; P = x2+y2-2zz clipped;
// rowmin = min over j (axis=2), colmin = min over i (axis=1).
__global__ __launch_bounds__(128)
void chamfer_wmma_kernel(const float* __restrict__ x, const float* __restrict__ y,
                         const float* __restrict__ x2, const float* __restrict__ y2,
                         float* __restrict__ rowmin, unsigned* __restrict__ colmin) {
    __shared__ float    sY[2][JT * YSTRIDE];  // double-buffered y tiles (async-filled)
    __shared__ unsigned sCol[NN];             // block-wide column mins (bit pattern)

    const int tid  = threadIdx.x;
    const int wave = tid >> 5;
    const int lane = tid & 31;
    const int m    = lane & 15;   // row (A) / col (B) index within 16
    const int half = lane >> 4;   // K-half select for 16x16x4 f32 fragments

    const int b  = blockIdx.x / (NN / ROWS_PER_BLOCK);
    const int i0 = (blockIdx.x % (NN / ROWS_PER_BLOCK)) * ROWS_PER_BLOCK + wave * 16;

    const float* xb = x + ((size_t)b * NN + i0) * ND;  // this wave's 16 rows
    const float* yb = y + (size_t)b * NN * ND;

    for (int j = tid; j < NN; j += 128) sCol[j] = CLIP_HI_BITS;

    // A fragments held in registers for the whole J loop (immune to asm clobbers):
    // lane (m,half), k-step s -> x[i0+m][4s+2*half], x[i0+m][4s+2*half+1]
    v2f xa[ND / 4];
    #pragma unroll
    for (int s = 0; s < ND / 4; ++s)
        xa[s] = *(const v2f*)(xb + (size_t)m * ND + 4 * s + 2 * half);

    // per-lane x^2 for the 8 rows this lane's accumulator VGPRs map to (M = v + 8*half)
    float x2r[8];
    #pragma unroll
    for (int v = 0; v < 8; ++v) x2r[v] = x2[(size_t)b * NN + i0 + v + 8 * half];

    float rmin[8];
    #pragma unroll
    for (int v = 0; v < 8; ++v) rmin[v] = 3.4e38f;

    // prologue: async-stage tile 0 into buffer 0
    stage_tile_async(&sY[0][0], yb, 0, tid);

    int cur = 0;
    for (int j0 = 0; j0 < NN; j0 += JT) {
        const int nxt = cur ^ 1;
        if (j0 + JT < NN) {
            // overlap: async-stage next tile into the other buffer...
            stage_tile_async(&sY[nxt][0], yb, j0 + JT, tid);
            // ...then wait only for the CURRENT tile's 8 loads (in-order completion)
            asm volatile("s_wait_asynccnt 0x8" ::: "memory");
            // L2 prefetch two tiles ahead (128B granules)
            if (j0 + 2 * JT < NN) {
                int row = tid >> 2, seg = tid & 3;
                __builtin_prefetch(yb + (size_t)(j0 + 2 * JT + row) * ND + seg * 32, 0, 1);
            }
        } else {
            asm volatile("s_wait_asynccnt 0x0" ::: "memory");
        }
        __syncthreads();  // publish current tile across all 4 waves

        const float* sYc = &sY[cur][0];
        v8f acc0 = {};   // columns j0 + 0..15
        v8f acc1 = {};   // columns j0 + 16..31
        #pragma unroll
        for (int s = 0; s < ND / 4; ++s) {
            const int k = 4 * s;
            // B fragments: lane (m,half) -> y[j0(+16)+m][k+2*half .. +1]
            v2f b0 = *(const v2f*)(sYc + (m)      * YSTRIDE + k + 2 * half);
            v2f b1 = *(const v2f*)(sYc + (m + 16) * YSTRIDE + k + 2 * half);
            acc0 = __builtin_amdgcn_wmma_f32_16x16x4_f32(false, xa[s], false, b0,
                                                         (short)0, acc0, false, false);
            acc1 = __builtin_amdgcn_wmma_f32_16x16x4_f32(false, xa[s], false, b1,
                                                         (short)0, acc1, false, false);
        }

        // ---- epilogue: P = x2 + y2 - 2*zz, clip, fold mins ----
        float y2a = y2[(size_t)b * NN + j0 + m];
        float y2b = y2[(size_t)b * NN + j0 + 16 + m];
        float cm0 = 3.4e38f, cm1 = 3.4e38f;
        #pragma unroll
        for (int v = 0; v < 8; ++v) {
            float p0 = fminf(fmaxf(x2r[v] + y2a - 2.0f * acc0[v], 0.0f), 100.0f);
            float p1 = fminf(fmaxf(x2r[v] + y2b - 2.0f * acc1[v], 0.0f), 100.0f);
            rmin[v] = fminf(rmin[v], fminf(p0, p1));   // same row, two columns
            cm0 = fminf(cm0, p0);
            cm1 = fminf(cm1, p1);
        }
        // combine the two halves holding the same column (lane n <-> lane n+16)
        cm0 = fminf(cm0, __shfl_xor(cm0, 16));
        cm1 = fminf(cm1, __shfl_xor(cm1, 16));
        if (half == 0) {
            atomicMin(&sCol[j0 + m],      __float_as_uint(cm0));  // P>=0: uint order == float order
            atomicMin(&sCol[j0 + 16 + m], __float_as_uint(cm1));
        }

        __syncthreads();  // all reads of sY[cur] done before iter i+1 async-overwrites it
        cur = nxt;
    }

    // ---- row mins: reduce across the 16 lanes of each half, store once ----
    #pragma unroll
    for (int v = 0; v < 8; ++v) {
        float r = rmin[v];
        r = fminf(r, __shfl_xor(r, 1));
        r = fminf(r, __shfl_xor(r, 2));
        r = fminf(r, __shfl_xor(r, 4));
        r = fminf(r, __shfl_xor(r, 8));
        rmin[v] = r;
    }
    if (m == 0) {
        #pragma unroll
        for (int v = 0; v < 8; ++v)
            rowmin[(size_t)b * NN + i0 + v + 8 * half] = rmin[v];
    }

    __syncthreads();
    for (int j = tid; j < NN; j += 128)
        atomicMin(&colmin[(size_t)b * NN + j], sCol[j]);
}

// ---------------- final reduction: (mean colmin + mean rowmin) / B ----------------
__global__ void final_reduce_kernel(const float* __restrict__ rowmin,
                                    const unsigned* __restrict__ colmin,
                                    float* __restrict__ out, int n) {
    __shared__ float sh[256];
    float s = 0.0f;
    for (int i = threadIdx.x; i < n; i += 256)
        s += rowmin[i] + __uint_as_float(colmin[i]);
    sh[threadIdx.x] = s;
    __syncthreads();
    for (int off = 128; off > 0; off >>= 1) {
        if (threadIdx.x < off) sh[threadIdx.x] += sh[threadIdx.x + off];
        __syncthreads();
    }
    // (sumCol/(B*N) + sumRow/(B*N)) * LOSS_WEIGHT / B == (sumCol+sumRow)/65536
    if (threadIdx.x == 0) out[0] = sh[0] * (1.0f / ((float)NB * (float)NN * (float)NB));
}

extern "C" void kernel_launch(void* const* d_in, const int* in_sizes, int n_in,
                              void* d_out, int out_size, void* d_ws, size_t ws_size,
                              hipStream_t stream) {
    const float* x = (const float*)d_in[0];  // corr_pred
    const float* y = (const float*)d_in[1];  // corr_target
    float* out = (float*)d_out;

    const int rows = NB * NN;  // 16384
    float*    x2     = (float*)d_ws;
    float*    y2     = x2 + rows;
    float*    rowmin = y2 + rows;
    unsigned* colmin = (unsigned*)(rowmin + rows);

    row_norms_kernel<<<rows / 256, 256, 0, stream>>>(x, x2, rows, ND);
    row_norms_kernel<<<rows / 256, 256, 0, stream>>>(y, y2, rows, ND);
    init_colmin_kernel<<<rows / 256, 256, 0, stream>>>(colmin, rows);
    chamfer_wmma_kernel<<<NB * (NN / ROWS_PER_BLOCK), 128, 0, stream>>>(
        x, y, x2, y2, rowmin, colmin);
    final_reduce_kernel<<<1, 256, 0, stream>>>(rowmin, colmin, out, rows);
}